// TriDistance_51960514347808
// MI455X (gfx1250) — compile-verified
//
#include <hip/hip_runtime.h>
#include <hip/hip_bf16.h>
#include <stdint.h>

typedef __attribute__((ext_vector_type(2))) float v2f;
typedef __attribute__((ext_vector_type(8))) float v8f;

#define B_DIM  4
#define N_PTS  4096
#define M_TRI  512
#define TILE   16   // points per wave (WMMA M)
#define CHUNK  16   // triangles per WMMA (WMMA N)

__device__ __forceinline__ float dot3(float ax, float ay, float az,
                                      float bx, float by, float bz) {
    return fmaf(ax, bx, fmaf(ay, by, az * bz));
}
__device__ __forceinline__ float safediv(float x, float y) {
    float yy = (fabsf(y) < 1e-12f) ? 1.0f : y;
    return x / yy;
}
__device__ __forceinline__ float clamp01(float x) {
    return fminf(fmaxf(x, 0.0f), 1.0f);
}

__global__ __launch_bounds__(32)
void tri_dist_wmma_kernel(const float* __restrict__ xyz1,
                          const float* __restrict__ tri1,
                          const float* __restrict__ tri2,
                          const float* __restrict__ tri3,
                          float* __restrict__ out_dist,
                          int*   __restrict__ out_region,
                          int*   __restrict__ out_index) {
    const int lane  = threadIdx.x;            // 0..31, wave32
    const int tile  = blockIdx.x;             // 0 .. B*N/16-1
    const int batch = tile / (N_PTS / TILE);
    const int tbase = (tile % (N_PTS / TILE)) * TILE;
    const int half  = lane >> 4;              // 0: rows 0..7, 1: rows 8..15
    const int l16   = lane & 15;

    // ---- load this lane's point (lanes L and L+16 hold point l16) ----
    const float* pp = xyz1 + ((size_t)batch * N_PTS + tbase + l16) * 3;
    const float px = pp[0], py = pp[1], pz = pp[2];
    const float np_self = dot3(px, py, pz, px, py, pz);

    // A fragment, 16x4 f32: lanes 0-15 -> (K0,K1)=(px,py); lanes 16-31 -> (K2,K3)=(pz,0)
    v2f afrag;
    afrag.x = half ? pz : px;
    afrag.y = half ? 0.0f : py;

    unsigned long long best[8];
#pragma unroll
    for (int r = 0; r < 8; ++r) best[r] = ~0ull;

    const float* t1b = tri1 + (size_t)batch * M_TRI * 3;
    const float* t2b = tri2 + (size_t)batch * M_TRI * 3;
    const float* t3b = tri3 + (size_t)batch * M_TRI * 3;

    for (int chunk = 0; chunk < M_TRI; chunk += CHUNK) {
        const int j = chunk + l16;            // this lane's triangle (column N = l16)
        const float* ap  = t1b + (size_t)j * 3;
        const float* bp_ = t2b + (size_t)j * 3;
        const float* cp_ = t3b + (size_t)j * 3;
        const float ax = ap[0],  ay = ap[1],  az = ap[2];
        const float bx = bp_[0], by = bp_[1], bz = bp_[2];
        const float cx = cp_[0], cy = cp_[1], cz = cp_[2];

        if (chunk + CHUNK < M_TRI) {          // gfx1250 global_prefetch_b8 path
            __builtin_prefetch(t1b + (size_t)(j + CHUNK) * 3, 0, 0);
            __builtin_prefetch(t2b + (size_t)(j + CHUNK) * 3, 0, 0);
            __builtin_prefetch(t3b + (size_t)(j + CHUNK) * 3, 0, 0);
        }

        // B fragments, 4x16 f32 (mirror of A layout, column N = l16)
        v2f bfA, bfB, bfC;
        bfA.x = half ? az : ax;  bfA.y = half ? 0.0f : ay;
        bfB.x = half ? bz : bx;  bfB.y = half ? 0.0f : by;
        bfC.x = half ? cz : cx;  bfC.y = half ? 0.0f : cy;

        v8f Sa = {}, Sb = {}, Sc = {};
        Sa = __builtin_amdgcn_wmma_f32_16x16x4_f32(false, afrag, false, bfA,
                                                   (short)0, Sa, false, false);
        Sb = __builtin_amdgcn_wmma_f32_16x16x4_f32(false, afrag, false, bfB,
                                                   (short)0, Sb, false, false);
        Sc = __builtin_amdgcn_wmma_f32_16x16x4_f32(false, afrag, false, bfC,
                                                   (short)0, Sc, false, false);

        // ---- per-lane (per-triangle) constants, shared by the 8 rows ----
        const float abx = bx - ax, aby = by - ay, abz = bz - az;
        const float acx = cx - ax, acy = cy - ay, acz = cz - az;
        const float cbx = cx - bx, cby = cy - by, cbz = cz - bz;
        const float k1  = dot3(abx, aby, abz, ax, ay, az);   // ab.a
        const float k3  = dot3(abx, aby, abz, bx, by, bz);   // ab.b
        const float k5  = dot3(abx, aby, abz, cx, cy, cz);   // ab.c
        const float k2  = dot3(acx, acy, acz, ax, ay, az);   // ac.a
        const float k4  = dot3(acx, acy, acz, bx, by, bz);   // ac.b
        const float k6  = dot3(acx, acy, acz, cx, cy, cz);   // ac.c
        const float kcb = dot3(cbx, cby, cbz, bx, by, bz);   // cb.b
        const float naa = dot3(ax, ay, az, ax, ay, az);
        const float nbb = dot3(bx, by, bz, bx, by, bz);
        const float ncc = dot3(cx, cy, cz, cx, cy, cz);
        const float nab2 = dot3(abx, aby, abz, abx, aby, abz);
        const float nac2 = dot3(acx, acy, acz, acx, acy, acz);
        const float abac = dot3(abx, aby, abz, acx, acy, acz);
        const float ncb2 = dot3(cbx, cby, cbz, cbx, cby, cbz);

#pragma unroll
        for (int r = 0; r < 8; ++r) {
            const float sa = Sa[r], sb = Sb[r], sc = Sc[r];
            // |p|^2 of row M = r + 8*half lives in lane (r + 8*half) of the low half
            const float np = __shfl(np_self, half ? (r + 8) : r, 32);

            const float u  = sb - sa;          // ab.p
            const float v  = sc - sa;          // ac.p
            const float d1 = u - k1, d3 = u - k3, d5 = u - k5;
            const float d2 = v - k2, d4 = v - k4, d6 = v - k6;
            const float vc = d1 * d4 - d3 * d2;
            const float vb = d5 * d2 - d1 * d6;
            const float va = d3 * d6 - d5 * d4;
            const float npa = np - 2.0f * sa + naa;   // |p-a|^2
            const float npb = np - 2.0f * sb + nbb;   // |p-b|^2
            const float npc = np - 2.0f * sc + ncc;   // |p-c|^2

            // default: face
            const float denom = va + vb + vc;
            const float vf = safediv(vb, denom);
            const float wf = safediv(vc, denom);
            float dist2 = npa - 2.0f * vf * d1 - 2.0f * wf * d2
                        + vf * vf * nab2 + wf * wf * nac2 + 2.0f * vf * wf * abac;
            int region = 6;

            // selection in reference order (later overrides earlier)
            const float tbc  = clamp01(safediv(d4 - d3, (d4 - d3) + (d5 - d6)));
            const float cbbp = (sc - sb) - kcb;       // cb.(p-b)
            const bool cbc = (va <= 0.0f) && ((d4 - d3) >= 0.0f) && ((d5 - d6) >= 0.0f);
            const float dbc = npb - 2.0f * tbc * cbbp + tbc * tbc * ncb2;
            dist2 = cbc ? dbc : dist2;  region = cbc ? 5 : region;

            const float tac = clamp01(safediv(d2, d2 - d6));
            const bool cac = (vb <= 0.0f) && (d2 >= 0.0f) && (d6 <= 0.0f);
            const float dac = npa - 2.0f * tac * d2 + tac * tac * nac2;
            dist2 = cac ? dac : dist2;  region = cac ? 4 : region;

            const bool cC = (d6 >= 0.0f) && (d5 <= d6);
            dist2 = cC ? npc : dist2;   region = cC ? 2 : region;

            const float tab = clamp01(safediv(d1, d1 - d3));
            const bool cab = (vc <= 0.0f) && (d1 >= 0.0f) && (d3 <= 0.0f);
            const float dab = npa - 2.0f * tab * d1 + tab * tab * nab2;
            dist2 = cab ? dab : dist2;  region = cab ? 3 : region;

            const bool cB = (d3 >= 0.0f) && (d4 <= d3);
            dist2 = cB ? npb : dist2;   region = cB ? 1 : region;

            const bool cA = (d1 <= 0.0f) && (d2 <= 0.0f);
            dist2 = cA ? npa : dist2;   region = cA ? 0 : region;

            dist2 = fmaxf(dist2, 0.0f);
            const unsigned long long key =
                ((unsigned long long)__float_as_uint(dist2) << 32) |
                (unsigned)(j * 8 + region);
            best[r] = (key < best[r]) ? key : best[r];
        }
    }

    // ---- min-reduce each row over its 16-lane half (xor stays within halves) ----
#pragma unroll
    for (int r = 0; r < 8; ++r) {
        unsigned long long k = best[r];
#pragma unroll
        for (int m = 1; m < 16; m <<= 1) {
            unsigned long long o = __shfl_xor(k, m, 32);
            k = (o < k) ? o : k;
        }
        best[r] = k;
    }

    // lane 0 writes rows 0..7, lane 16 writes rows 8..15
    if (l16 == 0) {
#pragma unroll
        for (int r = 0; r < 8; ++r) {
            const int row = r + half * 8;
            const size_t g = (size_t)batch * N_PTS + tbase + row;
            const unsigned long long k = best[r];
            out_dist[g]   = __uint_as_float((unsigned)(k >> 32));
            const unsigned low = (unsigned)(k & 0xffffffffu);
            out_region[g] = (int)(low & 7u);
            out_index[g]  = (int)(low >> 3);
        }
    }
}

extern "C" void kernel_launch(void* const* d_in, const int* in_sizes, int n_in,
                              void* d_out, int out_size, void* d_ws, size_t ws_size,
                              hipStream_t stream) {
    (void)in_sizes; (void)n_in; (void)out_size; (void)d_ws; (void)ws_size;
    const float* xyz1 = (const float*)d_in[0];
    const float* tri1 = (const float*)d_in[1];
    const float* tri2 = (const float*)d_in[2];
    const float* tri3 = (const float*)d_in[3];

    const size_t BN = (size_t)B_DIM * N_PTS;          // 16384 points total
    float* out_dist  = (float*)d_out;                 // [BN] f32
    int*   out_region = (int*)d_out + BN;             // [BN] i32
    int*   out_index  = (int*)d_out + 2 * BN;         // [BN] i32

    const int blocks = (B_DIM * N_PTS) / TILE;        // 1024 waves, one per block
    tri_dist_wmma_kernel<<<blocks, 32, 0, stream>>>(
        xyz1, tri1, tri2, tri3, out_dist, out_region, out_index);
}